// MaxSim_89764816487051
// MI455X (gfx1250) — compile-verified
//
#include <hip/hip_runtime.h>
#include <math.h>

typedef __attribute__((ext_vector_type(2))) float v2f;
typedef __attribute__((ext_vector_type(8))) float v8f;

#define Q      32
#define DLEN   2048
#define KDIM   128
#define KP     132                 // padded LDS row stride in floats (bank-conflict free)
#define NWAVE  8
#define TILES_PER_WAVE (DLEN / 16 / NWAVE)   // 16
#define EPSN   1e-12f

// LDS layout (in floats)
#define SM_QBUF   0                              // Q * KP
#define SM_DBUF   (SM_QBUF + Q * KP)             // NWAVE * 16 * KP
#define SM_DNORM  (SM_DBUF + NWAVE * 16 * KP)    // NWAVE * 16
#define SM_RQ     (SM_DNORM + NWAVE * 16)        // Q
#define SM_QPART  (SM_RQ + Q)                    // Q * 8
#define SM_WMAX   (SM_QPART + Q * 8)             // NWAVE * Q
#define SM_TOTALF (SM_WMAX + NWAVE * Q)

__global__ void __launch_bounds__(256)
maxsim_wmma_kernel(const float* __restrict__ qg,
                   const float* __restrict__ dg,
                   float* __restrict__ out)
{
    extern __shared__ float sm[];
    float* qbuf  = sm + SM_QBUF;
    float* dbufA = sm + SM_DBUF;
    float* dnrmA = sm + SM_DNORM;
    float* rq    = sm + SM_RQ;
    float* qpart = sm + SM_QPART;
    float* wmax  = sm + SM_WMAX;

    const int b    = blockIdx.x;
    const int tid  = threadIdx.x;
    const int lane = tid & 31;
    const int wave = tid >> 5;

    // ---------------- 1) load + L2-normalize queries into LDS ----------------
    {
        const int row = tid >> 3;          // 0..31
        const int seg = tid & 7;           // 16 floats each
        const float4* src = (const float4*)(qg + ((size_t)b * Q + row) * KDIM + seg * 16);
        float4 v[4];
        float ss = 0.f;
#pragma unroll
        for (int j = 0; j < 4; ++j) {
            v[j] = src[j];
            ss += v[j].x * v[j].x + v[j].y * v[j].y + v[j].z * v[j].z + v[j].w * v[j].w;
        }
#pragma unroll
        for (int j = 0; j < 4; ++j)
            *(float4*)&qbuf[row * KP + seg * 16 + j * 4] = v[j];
        qpart[row * 8 + seg] = ss;
    }
    __syncthreads();
    if (tid < Q) {
        float ss = 0.f;
#pragma unroll
        for (int j = 0; j < 8; ++j) ss += qpart[tid * 8 + j];
        rq[tid] = 1.f / fmaxf(sqrtf(ss), EPSN);
    }
    __syncthreads();
    {
        const int row = tid >> 3, seg = tid & 7;
        const float r = rq[row];
#pragma unroll
        for (int j = 0; j < 16; ++j) qbuf[row * KP + seg * 16 + j] *= r;
    }
    __syncthreads();

    // ---------------- 2) per-wave resident A fragments (2 query tiles) -------
    const int m    = lane & 15;            // row (A) / column (B,C)
    const int koff = (lane >> 4) * 2;      // K sub-offset within a K=4 step
    v2f afrag[2][32];
#pragma unroll
    for (int t = 0; t < 2; ++t)
#pragma unroll
        for (int k = 0; k < 32; ++k)
            afrag[t][k] = *(v2f*)&qbuf[(t * 16 + m) * KP + k * 4 + koff];

    // ---------------- 3) stream doc tiles, WMMA, running max -----------------
    float* db = dbufA + wave * 16 * KP;
    float* dn = dnrmA + wave * 16;
    float vmax0[8], vmax1[8];
#pragma unroll
    for (int v = 0; v < 8; ++v) { vmax0[v] = -INFINITY; vmax1[v] = -INFINITY; }

    for (int i = 0; i < TILES_PER_WAVE; ++i) {
        const int tile = i * NWAVE + wave;
        const float* base = dg + ((size_t)b * DLEN + (size_t)tile * 16) * KDIM;

        // stage 16 docs x 128 dims into LDS, fully coalesced, compute 1/||d||
#pragma unroll
        for (int r = 0; r < 16; ++r) {
            float4 f = ((const float4*)(base + r * KDIM))[lane];
            *(float4*)&db[r * KP + lane * 4] = f;
            float p = f.x * f.x + f.y * f.y + f.z * f.z + f.w * f.w;
            p += __shfl_xor(p, 16); p += __shfl_xor(p, 8); p += __shfl_xor(p, 4);
            p += __shfl_xor(p, 2);  p += __shfl_xor(p, 1);
            if (lane == 0) dn[r] = 1.f / fmaxf(sqrtf(p), EPSN);
            if (i + 1 < TILES_PER_WAVE)
                __builtin_prefetch(base + (size_t)NWAVE * 16 * KDIM + r * KDIM + lane * 4, 0, 1);
        }

        // 16(q) x 16(doc) x 128(k) via 32 K-steps of V_WMMA_F32_16X16X4_F32,
        // two query tiles share each B fragment
        v8f c0 = {}; v8f c1 = {};
#pragma unroll
        for (int k = 0; k < 32; ++k) {
            v2f bf = *(v2f*)&db[m * KP + k * 4 + koff];
            c0 = __builtin_amdgcn_wmma_f32_16x16x4_f32(false, afrag[0][k], false, bf,
                                                       (short)0, c0, false, false);
            c1 = __builtin_amdgcn_wmma_f32_16x16x4_f32(false, afrag[1][k], false, bf,
                                                       (short)0, c1, false, false);
        }

        // fold doc norm (per column) and update running max
        const float rn = dn[m];
#pragma unroll
        for (int v = 0; v < 8; ++v) {
            vmax0[v] = fmaxf(vmax0[v], c0[v] * rn);
            vmax1[v] = fmaxf(vmax1[v], c1[v] * rn);
        }
    }

    // ---------------- 4) reduce max over the 16 column lanes -----------------
#pragma unroll
    for (int v = 0; v < 8; ++v) {
#pragma unroll
        for (int s = 1; s <= 8; s <<= 1) {
            vmax0[v] = fmaxf(vmax0[v], __shfl_xor(vmax0[v], s));
            vmax1[v] = fmaxf(vmax1[v], __shfl_xor(vmax1[v], s));
        }
    }
    if (m == 0) {
        const int half = (lane >> 4) * 8;   // rows 0-7 vs 8-15 of each tile
#pragma unroll
        for (int v = 0; v < 8; ++v) {
            wmax[wave * Q + half + v]      = vmax0[v];
            wmax[wave * Q + 16 + half + v] = vmax1[v];
        }
    }
    __syncthreads();

    // ---------------- 5) cross-wave max per query, then sum ------------------
    if (tid < Q) {
        float mx = -INFINITY;
#pragma unroll
        for (int w = 0; w < NWAVE; ++w) mx = fmaxf(mx, wmax[w * Q + tid]);
        mx += __shfl_xor(mx, 1); mx += __shfl_xor(mx, 2); mx += __shfl_xor(mx, 4);
        mx += __shfl_xor(mx, 8); mx += __shfl_xor(mx, 16);
        if (tid == 0) out[b] = mx;
    }
}

extern "C" void kernel_launch(void* const* d_in, const int* in_sizes, int n_in,
                              void* d_out, int out_size, void* d_ws, size_t ws_size,
                              hipStream_t stream)
{
    (void)n_in; (void)d_ws; (void)ws_size; (void)out_size;
    const float* qg = (const float*)d_in[0];
    const float* dg = (const float*)d_in[1];
    float* out = (float*)d_out;

    const int B = in_sizes[1] / (DLEN * KDIM);   // 1024 for the reference shapes
    const size_t smem = (size_t)SM_TOTALF * sizeof(float);
    maxsim_wmma_kernel<<<dim3(B), dim3(256), smem, stream>>>(qg, dg, out);
}